// SortNet_377957122201
// MI455X (gfx1250) — compile-verified
//
#include <hip/hip_runtime.h>
#include <stdint.h>

#define B_    16
#define F_    64
#define N_    32768
#define C0_   64
#define C1_   16
#define TOPK_ 64

typedef __attribute__((ext_vector_type(2))) float v2f;
typedef __attribute__((ext_vector_type(8))) float v8f;

// Workspace layout (floats):
//   [0..63]    s0_sum      [64..127]  s0_sq
//   [128..191] bn0_scale   [192..255] bn0_shift
//   [256..271] s1_sum      [272..287] s1_sq
//   [288..303] bn1_scale   [304..319] bn1_shift
//   [320] s2_sum  [321] s2_sq  [322] bn2_scale  [323] bn2_shift
//   [512 ..)   x1  (B*C1*N floats = 32 MB)
//   [512+B*C1*N ..) x2 (B*N floats = 2 MB)
#define WS_X1   512
#define WS_X2   (512 + (size_t)B_ * C1_ * N_)

// Pair-interleaved LDS tiles: element (f, n) lives at [f>>1][n][f&1].
// Pair-row stride = 80 pairs = 160 dwords == 32 (mod 64 banks): the two
// half-wave groups (rows 2kk / 2kk+1) hit disjoint bank halves -> conflict-free b64.
#define PPAD 80

__device__ __forceinline__ v8f wmma4(v2f a, v2f b, v8f c) {
  // V_WMMA_F32_16X16X4_F32: D = A(16x4) * B(4x16) + C(16x16)
  return __builtin_amdgcn_wmma_f32_16x16x4_f32(
      /*neg_a=*/false, a, /*neg_b=*/false, b,
      /*c_mod=*/(short)0, c, /*reuse_a=*/false, /*reuse_b=*/false);
}

// CDNA5 async global->LDS scatter (ASYNCcnt-tracked, per-lane LDS dest).
__device__ __forceinline__ void async_load_b32(unsigned lds_byte, const float* gaddr) {
  asm volatile("global_load_async_to_lds_b32 %0, %1, off"
               :: "v"(lds_byte), "v"(gaddr)
               : "memory");
}
__device__ __forceinline__ void wait_asynccnt0() {
  asm volatile("s_wait_asynccnt 0" ::: "memory");
}

// GEMM0: async-scatter the 64x64 sortvec tile into pair-interleaved LDS once per
// WG, then acc[nt] = w0[wave*16..+16, 0..64) x tile[0..64, nt*16..+16).
// Hot loop = one ds_load_b64 + one v_wmma per step (operand lands pre-paired).
__device__ __forceinline__ void gemm0_tile(const float* __restrict__ svb,
                                           const float* __restrict__ w0,
                                           float* tilep,   // [32][PPAD][2]
                                           int n0, int wave, int lane, v8f acc[4]) {
  const int half = lane >> 4;
  const int lcol = lane & 15;
  const int mrow = wave * 16 + lcol;

  // Stage: wave owns rows f in [wave*16, wave*16+16); 2 async b32 per row
  // (32 lanes cover 64 columns).
  const unsigned lds_base = (unsigned)(uintptr_t)tilep;
#pragma unroll
  for (int rr = 0; rr < 16; ++rr) {
    const int f = wave * 16 + rr;
#pragma unroll
    for (int cc = 0; cc < 2; ++cc) {
      const int n = cc * 32 + lane;
      const float* g = svb + (size_t)f * N_ + n0 + n;
      const unsigned l = lds_base +
          (unsigned)((((f >> 1) * PPAD + n) * 2 + (f & 1)) * 4);
      async_load_b32(l, g);
    }
  }

  // Preload A (w0 tile) while the DMA is in flight.
  v2f A[16];
#pragma unroll
  for (int kk = 0; kk < 16; ++kk) {
    A[kk].x = w0[mrow * F_ + kk * 4 + 2 * half];
    A[kk].y = w0[mrow * F_ + kk * 4 + 2 * half + 1];
  }

  wait_asynccnt0();
  __syncthreads();

  v8f z = {0.f, 0.f, 0.f, 0.f, 0.f, 0.f, 0.f, 0.f};
#pragma unroll
  for (int nt = 0; nt < 4; ++nt) acc[nt] = z;
#pragma unroll
  for (int kk = 0; kk < 16; ++kk) {
    const int p = kk * 2 + half;     // pair row holding (4kk+2h, 4kk+2h+1)
#pragma unroll
    for (int nt = 0; nt < 4; ++nt) {
      const int n = nt * 16 + lcol;
      v2f bv = *(const v2f*)&tilep[(p * PPAD + n) * 2];   // ds_load_b64
      acc[nt] = wmma4(A[kk], bv, acc[nt]);
    }
  }
}

// Pass 1: GEMM0 + per-channel sum/sumsq (conv bias cancels in BN, skipped).
__global__ void __launch_bounds__(128)
k_gemm0_stats(const float* __restrict__ sv, const float* __restrict__ w0,
              float* __restrict__ ws) {
  __shared__ float tilep[32 * PPAD * 2];
  const int wave = threadIdx.x >> 5, lane = threadIdx.x & 31;
  const int half = lane >> 4, lcol = lane & 15;
  const int b = blockIdx.y, n0 = blockIdx.x * 64;
  const float* svb = sv + (size_t)b * F_ * N_;
  v8f acc[4];
  gemm0_tile(svb, w0, tilep, n0, wave, lane, acc);
#pragma unroll
  for (int r = 0; r < 8; ++r) {
    float s = acc[0][r] + acc[1][r] + acc[2][r] + acc[3][r];
    float q = acc[0][r] * acc[0][r] + acc[1][r] * acc[1][r] +
              acc[2][r] * acc[2][r] + acc[3][r] * acc[3][r];
#pragma unroll
    for (int m = 1; m < 16; m <<= 1) {
      s += __shfl_xor(s, m, 32);
      q += __shfl_xor(q, m, 32);
    }
    if (lcol == 0) {
      int ch = wave * 16 + r + 8 * half;
      atomicAdd(&ws[ch], s);
      atomicAdd(&ws[64 + ch], q);
    }
  }
}

// Generic BN finalize: scale = g*rsqrt(var+eps), shift = bt - mean*scale
__global__ void k_finalize(const float* __restrict__ sum, const float* __restrict__ sq,
                           const float* __restrict__ g, const float* __restrict__ bt,
                           float* __restrict__ scale, float* __restrict__ shift,
                           int C, float inv_count) {
  int c = threadIdx.x;
  if (c < C) {
    float mean = sum[c] * inv_count;
    float var = sq[c] * inv_count - mean * mean;
    float is = rsqrtf(var + 1e-5f);
    float s = g[c] * is;
    scale[c] = s;
    shift[c] = bt[c] - mean * s;
  }
}

// Pass 2: recompute GEMM0 (L2-resident), BN0+ReLU -> pair-interleaved LDS,
// WMMA GEMM1 + stats, store x1.
__global__ void __launch_bounds__(128)
k_gemm0_bn_gemm1(const float* __restrict__ sv, const float* __restrict__ w0,
                 const float* __restrict__ w1, float* __restrict__ ws) {
  __shared__ float tilep[32 * PPAD * 2];
  __shared__ float ytp[32 * PPAD * 2];   // y0 pairs: [f>>1][n][f&1]
  const int wave = threadIdx.x >> 5, lane = threadIdx.x & 31;
  const int half = lane >> 4, lcol = lane & 15;
  const int b = blockIdx.y, n0 = blockIdx.x * 64;
  const float* svb = sv + (size_t)b * F_ * N_;
  v8f acc[4];
  gemm0_tile(svb, w0, tilep, n0, wave, lane, acc);

  const float* sc0 = ws + 128;
  const float* sh0 = ws + 192;
#pragma unroll
  for (int r = 0; r < 8; r += 2) {
    const int m = wave * 16 + r + 8 * half;      // even channel
    const float sa = sc0[m], ha = sh0[m];
    const float sb = sc0[m + 1], hb = sh0[m + 1];
#pragma unroll
    for (int nt = 0; nt < 4; ++nt) {
      v2f p;
      p.x = fmaxf(fmaf(acc[nt][r], sa, ha), 0.f);
      p.y = fmaxf(fmaf(acc[nt][r + 1], sb, hb), 0.f);
      *(v2f*)&ytp[((m >> 1) * PPAD + nt * 16 + lcol) * 2] = p;  // ds_store_b64
    }
  }
  __syncthreads();

  // GEMM1: x1[0..16, n] = w1[16x64] * y0[64 x 16]; each wave owns nt = wave.
  v8f acc1 = {0.f, 0.f, 0.f, 0.f, 0.f, 0.f, 0.f, 0.f};
#pragma unroll
  for (int kk = 0; kk < 16; ++kk) {
    v2f a;
    a.x = w1[lcol * 64 + kk * 4 + 2 * half];
    a.y = w1[lcol * 64 + kk * 4 + 2 * half + 1];
    v2f bv = *(const v2f*)&ytp[((kk * 2 + half) * PPAD + wave * 16 + lcol) * 2];
    acc1 = wmma4(a, bv, acc1);
  }

  float* x1 = ws + WS_X1;
  const int n = n0 + wave * 16 + lcol;
#pragma unroll
  for (int r = 0; r < 8; ++r) {
    int m1 = r + 8 * half;
    float v = acc1[r];
    x1[((size_t)(b * C1_ + m1)) * N_ + n] = v;
    float s = v, q = v * v;
#pragma unroll
    for (int m = 1; m < 16; m <<= 1) {
      s += __shfl_xor(s, m, 32);
      q += __shfl_xor(q, m, 32);
    }
    if (lcol == 0) {
      atomicAdd(&ws[256 + m1], s);
      atomicAdd(&ws[272 + m1], q);
    }
  }
}

// Pass 3: BN1+ReLU, 16-wide dot with w2 -> x2, stats for BN2.
__global__ void __launch_bounds__(256)
k_stage2(const float* __restrict__ w2, float* __restrict__ ws) {
  const int g = blockIdx.x * 256 + threadIdx.x;
  const int b = g >> 15, n = g & (N_ - 1);
  const float* x1 = ws + WS_X1;
  const float* sc1 = ws + 288;
  const float* sh1 = ws + 304;
  float acc = 0.f;
#pragma unroll
  for (int c = 0; c < C1_; ++c) {
    float v = x1[((size_t)(b * C1_ + c)) * N_ + n];
    v = fmaxf(fmaf(v, sc1[c], sh1[c]), 0.f);
    acc = fmaf(w2[c], v, acc);
  }
  float* x2 = ws + WS_X2;
  x2[g] = acc;
  float s = acc, q = acc * acc;
#pragma unroll
  for (int m = 1; m < 32; m <<= 1) {
    s += __shfl_xor(s, m, 32);
    q += __shfl_xor(q, m, 32);
  }
  if ((threadIdx.x & 31) == 0) {
    atomicAdd(&ws[320], s);
    atomicAdd(&ws[321], q);
  }
}

// Pass 4: per-batch top-64 (scores staged in LDS; 320KB/WGP on CDNA5), gather.
__global__ void __launch_bounds__(256)
k_topk(const float* __restrict__ input, const float* __restrict__ ws,
       float* __restrict__ out) {
  extern __shared__ float smem[];
  float* sc = smem;                   // N_ scores
  float* rv = smem + N_;              // 256 reduce vals
  int*   ri = (int*)(rv + 256);       // 256 reduce idx
  float* wv = (float*)(ri + 256);     // 64 winner vals
  int*   wi = (int*)(wv + 64);        // 64 winner idx
  const int b = blockIdx.x, tid = threadIdx.x;
  const float* x2 = ws + WS_X2 + (size_t)b * N_;
  const float scale2 = ws[322], shift2 = ws[323];

  for (int n = tid; n < N_; n += 256)
    sc[n] = fmaxf(fmaf(x2[n], scale2, shift2), 0.f);
  __syncthreads();

  for (int k = 0; k < TOPK_; ++k) {
    float bv = -1.f;
    int bi = N_;
    for (int n = tid; n < N_; n += 256) {
      float v = sc[n];
      if (v > bv) { bv = v; bi = n; }  // strict > keeps lowest index on ties
    }
    rv[tid] = bv;
    ri[tid] = bi;
    __syncthreads();
    for (int s = 128; s > 0; s >>= 1) {
      if (tid < s) {
        float ov = rv[tid + s];
        int oi = ri[tid + s];
        if (ov > rv[tid] || (ov == rv[tid] && oi < ri[tid])) {
          rv[tid] = ov;
          ri[tid] = oi;
        }
      }
      __syncthreads();
    }
    if (tid == 0) {
      wv[k] = rv[0];
      wi[k] = ri[0];
      sc[ri[0]] = -2.f;  // remove winner
    }
    __syncthreads();
  }

  // feat[b, f, k]: f<64 -> gathered input column; f==64 -> score value.
  for (int t = tid; t < 65 * 64; t += 256) {
    int f = t >> 6, k = t & 63;
    float v = (f < 64) ? input[((size_t)(b * F_ + f)) * N_ + wi[k]] : wv[k];
    out[(size_t)(b * 65 + f) * 64 + k] = v;
  }
  int* oidx = (int*)(out + (size_t)B_ * 65 * 64);
  for (int t = tid; t < TOPK_; t += 256) oidx[b * 64 + t] = wi[t];
}

__global__ void k_init(float* __restrict__ ws) {
  if (threadIdx.x < 512) ws[threadIdx.x] = 0.f;
}

extern "C" void kernel_launch(void* const* d_in, const int* in_sizes, int n_in,
                              void* d_out, int out_size, void* d_ws, size_t ws_size,
                              hipStream_t stream) {
  const float* sv    = (const float*)d_in[0];
  const float* input = (const float*)d_in[1];
  const float* w0    = (const float*)d_in[2];
  const float* w1    = (const float*)d_in[4];
  const float* w2    = (const float*)d_in[6];
  const float* g0    = (const float*)d_in[8];
  const float* bt0   = (const float*)d_in[9];
  const float* g1    = (const float*)d_in[10];
  const float* bt1   = (const float*)d_in[11];
  const float* g2    = (const float*)d_in[12];
  const float* bt2   = (const float*)d_in[13];
  float* ws  = (float*)d_ws;
  float* out = (float*)d_out;

  const float invBN = 1.0f / ((float)B_ * (float)N_);
  dim3 gg(N_ / 64, B_);

  k_init<<<1, 512, 0, stream>>>(ws);
  k_gemm0_stats<<<gg, 128, 0, stream>>>(sv, w0, ws);
  k_finalize<<<1, 64, 0, stream>>>(ws + 0, ws + 64, g0, bt0, ws + 128, ws + 192, 64, invBN);
  k_gemm0_bn_gemm1<<<gg, 128, 0, stream>>>(sv, w0, w1, ws);
  k_finalize<<<1, 64, 0, stream>>>(ws + 256, ws + 272, g1, bt1, ws + 288, ws + 304, 16, invBN);
  k_stage2<<<(B_ * N_) / 256, 256, 0, stream>>>(w2, ws);
  k_finalize<<<1, 64, 0, stream>>>(ws + 320, ws + 321, g2, bt2, ws + 322, ws + 323, 1, invBN);

  size_t lds = (size_t)N_ * 4 + 256 * 4 + 256 * 4 + 64 * 4 + 64 * 4;
  k_topk<<<B_, 256, lds, stream>>>(input, ws, out);
}